// ALCOVECell_37520834297867
// MI455X (gfx1250) — compile-verified
//
#include <hip/hip_runtime.h>
#include <hip/hip_bf16.h>
#include <cstdint>

#define B_N 1024
#define R_N 512
#define D_N 128
#define O_N 64

#define BETA  6.5f
#define PHI   2.0f
#define LAM_A 0.0033f
#define LAM_W 0.003f

typedef float v2f __attribute__((ext_vector_type(2)));
typedef float v8f __attribute__((ext_vector_type(8)));
typedef unsigned int u32x4 __attribute__((ext_vector_type(4)));
typedef int i32x4 __attribute__((ext_vector_type(4)));
typedef int i32x8 __attribute__((ext_vector_type(8)));

// Toolchain-portable TDM selection:
//  - amdgpu-toolchain (clang-23 + therock headers) ships amd_gfx1250_TDM.h and
//    a 6-arg __builtin_amdgcn_tensor_load_to_lds
//  - ROCm 7.2 (clang-22) has the 5-arg form
#if defined(__AMDGCN__) && __has_builtin(__builtin_amdgcn_tensor_load_to_lds) && \
    __has_builtin(__builtin_amdgcn_s_wait_tensorcnt)
  #define USE_TDM 1
  #if __has_include(<hip/amd_detail/amd_gfx1250_TDM.h>)
    #define TDM_SIX_ARG 1
  #endif
#endif

static __device__ __forceinline__ v8f wmma_f32_k4(v2f a, v2f b, v8f c) {
    // D = A(16x4 f32) * B(4x16 f32) + C(16x16 f32)
    return __builtin_amdgcn_wmma_f32_16x16x4_f32(
        /*neg_a=*/false, a, /*neg_b=*/false, b,
        /*c_mod=*/(short)0, c, /*reuse_a=*/false, /*reuse_b=*/false);
}

// ---------------------------------------------------------------------------
// Stage 0: A_b[b] = sum_d attn[b,d] * z[b,d]^2   (one wave per sample)
// ---------------------------------------------------------------------------
__global__ __launch_bounds__(256) void alcove_rowsum(
    const float* __restrict__ attn, const float* __restrict__ z,
    float* __restrict__ Ab)
{
    int b    = blockIdx.x * 8 + (threadIdx.x >> 5);
    int lane = threadIdx.x & 31;
    const float* ar = attn + b * D_N;
    const float* zr = z    + b * D_N;
    float s = 0.f;
    #pragma unroll
    for (int k = lane; k < D_N; k += 32) {
        float zv = zr[k];
        s = fmaf(ar[k] * zv, zv, s);
    }
    #pragma unroll
    for (int off = 16; off > 0; off >>= 1) s += __shfl_down(s, off, 32);
    if (lane == 0) Ab[b] = s;
}

// ---------------------------------------------------------------------------
// Stage 1: d2 = A_b - 2*(attn.z)@N^T + attn@(N^2)^T  via f32 WMMA
//          s = exp(-BETA*sqrt(d2));  scale = BETA*s/(2d)
// One wave = one 16x16 (b,r) tile. K = D = 128, step 4.
// ---------------------------------------------------------------------------
__global__ __launch_bounds__(256) void alcove_rbf(
    const float* __restrict__ attn, const float* __restrict__ z,
    const float* __restrict__ rbf,  const float* __restrict__ Ab,
    float* __restrict__ sArr, float* __restrict__ scArr)
{
    int wave = threadIdx.x >> 5;
    int lane = threadIdx.x & 31;
    int tile = blockIdx.x * 8 + wave;
    int bm = tile >> 5;           // R/16 = 32 r-tiles per b-row
    int rn = tile & 31;
    int m  = lane & 15;
    int hi = lane >> 4;
    int kk = hi * 2;              // A lane supplies K = kk, kk+1
    const float* aRow = attn + (size_t)(bm * 16 + m) * D_N;
    const float* zRow = z    + (size_t)(bm * 16 + m) * D_N;
    const float* nRow = rbf  + (size_t)(rn * 16 + m) * D_N;   // B col n = lane&15

    v8f acc1 = {};   // (attn*z) @ N^T
    v8f acc2 = {};   // attn @ (N^2)^T
    for (int k0 = 0; k0 < D_N; k0 += 4) {
        v2f av = *(const v2f*)(aRow + k0 + kk);
        v2f zv = *(const v2f*)(zRow + k0 + kk);
        v2f nv = *(const v2f*)(nRow + k0 + kk);
        v2f aw = av * zv;
        v2f n2 = nv * nv;
        acc1 = wmma_f32_k4(aw, nv, acc1);
        acc2 = wmma_f32_k4(av, n2, acc2);
    }
    int col = rn * 16 + (lane & 15);
    #pragma unroll
    for (int j = 0; j < 8; ++j) {
        int row  = bm * 16 + j + hi * 8;
        float d2 = Ab[row] - 2.f * acc1[j] + acc2[j];
        d2 = fmaxf(d2, 0.f);
        float dd = sqrtf(d2);
        float sv = expf(-BETA * dd);
        sArr [row * R_N + col] = sv;
        scArr[row * R_N + col] = (dd > 1e-12f) ? (BETA * sv) / (2.f * dd) : 0.f;
    }
}

// ---------------------------------------------------------------------------
// Stage 2+3 fused, one block per sample b:
//   assoc[b] (512x64 f32 = 128KB) staged once in LDS via the Tensor Data
//   Mover. The D# pad fields (pad_interval=64 DWORDs, pad_amount=1 DWORD)
//   produce the stride-65 bank-conflict-free layout directly in LDS.
//   x_out -> teacher -> g ; h = g@assoc^T ; c = -scale*h ; Csum = sum c ;
//   new_assoc = assoc - LAM_W * g (x) s   (single read + single write of the
//   268MB-dominant tensor => HBM-bound ~11.5us floor at 23.3 TB/s)
// ---------------------------------------------------------------------------
#define RS 65
__global__ __launch_bounds__(256) void alcove_assoc(
    const float* __restrict__ assoc, const float* __restrict__ onehot,
    const float* __restrict__ sArr,  const float* __restrict__ scArr,
    float* __restrict__ xout, float* __restrict__ newAssoc,
    float* __restrict__ cArr, float* __restrict__ Csum)
{
    __shared__ float As[R_N * RS];   // 133,120 B (padded 512 x 65)
    __shared__ float sS[R_N];
    __shared__ float gS[O_N];
    __shared__ float red[256];
    const int b = blockIdx.x;
    const int t = threadIdx.x;

#if defined(USE_TDM)
    // ---- TDM: DMA assoc[b] (512 rows x 64 f32) global -> LDS, padded layout.
    if (t < 32) {                    // wave 0 issues the descriptor
        uint64_t gaddr = (uint64_t)(uintptr_t)(assoc + (size_t)b * R_N * O_N);
        uint32_t laddr = (uint32_t)(uintptr_t)(void*)&As[0];
        u32x4 g0;
        g0[0] = 1u;                                   // count=1, user descriptor
        g0[1] = laddr;                                // lds_addr
        g0[2] = (uint32_t)(gaddr & 0xFFFFFFFFu);      // global_addr[31:0]
        g0[3] = (uint32_t)((gaddr >> 32) & 0x01FFFFFFu) // global_addr[56:32]
              | 0x80000000u;                          // type=2 ("image")
        i32x8 g1;
        g1[0] = (2 << 16)      // data_size = 4B
              | (1 << 20)      // pad_enable
              | (5 << 22)      // pad_interval: 64 DWORDs
              | (0 << 25);     // pad_amount: 1 DWORD
        g1[1] = (O_N & 0xFFFF) << 16;                 // tensor_dim0[15:0] = 64
        g1[2] = (R_N & 0xFFFF) << 16;                 // dim0 hi=0 | tensor_dim1 lo = 512
        g1[3] = (O_N & 0xFFFF) << 16;                 // dim1 hi=0 | tile_dim0 = 64
        g1[4] = R_N;                                  // tile_dim1=512, tile_dim2=0
        g1[5] = O_N;                                  // tensor_dim0_stride = 64
        g1[6] = 0;
        g1[7] = 0;
        i32x4 gz = {0, 0, 0, 0};
#if defined(TDM_SIX_ARG)
        i32x8 gz8 = {0, 0, 0, 0, 0, 0, 0, 0};
        __builtin_amdgcn_tensor_load_to_lds(g0, g1, gz, gz, gz8, 0);
#else
        __builtin_amdgcn_tensor_load_to_lds(g0, g1, gz, gz, 0);
#endif
    }
    for (int i = t; i < R_N; i += 256) sS[i] = sArr[b * R_N + i]; // overlap DMA
    if (t < 32) __builtin_amdgcn_s_wait_tensorcnt(0);
    __syncthreads();
#else
    // ---- fallback: manual global->LDS staging
    {
        const float4* ab4 = (const float4*)(assoc + (size_t)b * R_N * O_N);
        #pragma unroll
        for (int k = 0; k < 32; ++k) {
            int i4 = t + k * 256;
            float4 v = ab4[i4];
            int i = i4 * 4;
            int r = i >> 6, o = i & 63;
            As[r * RS + o + 0] = v.x;  As[r * RS + o + 1] = v.y;
            As[r * RS + o + 2] = v.z;  As[r * RS + o + 3] = v.w;
        }
        for (int i = t; i < R_N; i += 256) sS[i] = sArr[b * R_N + i];
        __syncthreads();
    }
#endif

    // ---- forward: x_out[o] = sum_r s[r]*assoc[r,o] (4 partials per o)
    {
        int o = t & 63, chunk = t >> 6;
        float p = 0.f;
        int r0 = chunk * 128;
        for (int r = r0; r < r0 + 128; ++r) p = fmaf(sS[r], As[r * RS + o], p);
        red[t] = p;
    }
    __syncthreads();
    if (t < O_N) {
        float xo  = red[t] + red[t + 64] + red[t + 128] + red[t + 192];
        float lab = onehot[b * O_N + t];
        float tmin = fminf(-1.f, xo);
        float tmax = fmaxf( 1.f, xo);
        float teacher = tmin - lab * tmin + lab * tmax;
        xout[b * O_N + t] = PHI * xo;
        gS[t] = (2.0f / (float)O_N) * (xo - teacher);   // dL/dx_out
    }
    __syncthreads();

    // ---- h[r] = sum_o g[o]*assoc[r,o]; c = -scale*h; Csum
    float cpart = 0.f;
    #pragma unroll
    for (int rr = 0; rr < 2; ++rr) {
        int r = t + rr * 256;
        float h = 0.f;
        #pragma unroll 8
        for (int o = 0; o < O_N; ++o) h = fmaf(gS[o], As[r * RS + o], h);
        float c = -scArr[b * R_N + r] * h;
        cArr[b * R_N + r] = c;
        cpart += c;
    }
    red[t] = cpart;
    __syncthreads();
    for (int off = 128; off > 0; off >>= 1) {
        if (t < off) red[t] += red[t + off];
        __syncthreads();
    }
    if (t == 0) Csum[b] = red[0];

    // ---- new_assoc = assoc - LAM_W * g[o]*s[r] (stream out of LDS)
    float4* nb4 = (float4*)(newAssoc + (size_t)b * R_N * O_N);
    #pragma unroll
    for (int k = 0; k < 32; ++k) {
        int i4 = t + k * 256;
        int i = i4 * 4;
        int r = i >> 6, o = i & 63;
        float sr = sS[r];
        float4 v;
        v.x = As[r * RS + o + 0] - LAM_W * gS[o + 0] * sr;
        v.y = As[r * RS + o + 1] - LAM_W * gS[o + 1] * sr;
        v.z = As[r * RS + o + 2] - LAM_W * gS[o + 2] * sr;
        v.w = As[r * RS + o + 3] - LAM_W * gS[o + 3] * sr;
        nb4[i4] = v;
    }
}

// ---------------------------------------------------------------------------
// Stage 4: dl_da = z^2*Csum - 2z*(c@N) + c@N^2  via f32 WMMA (K = R = 512)
//          new_attn = max(attn - LAM_A*dl_da, 0)
// ---------------------------------------------------------------------------
__global__ __launch_bounds__(256) void alcove_attn(
    const float* __restrict__ cArr, const float* __restrict__ rbf,
    const float* __restrict__ z,    const float* __restrict__ attn,
    const float* __restrict__ Csum, float* __restrict__ newAttn)
{
    int wave = threadIdx.x >> 5;
    int lane = threadIdx.x & 31;
    int tile = blockIdx.x * 8 + wave;
    int bm = tile >> 3;           // D/16 = 8 d-tiles per b-row
    int dn = tile & 7;
    int m  = lane & 15;
    int hi = lane >> 4;
    int kk = hi * 2;
    int dcol = dn * 16 + (lane & 15);
    const float* cRow = cArr + (size_t)(bm * 16 + m) * R_N;

    v8f acc1 = {};   // c @ N
    v8f acc2 = {};   // c @ N^2
    for (int k0 = 0; k0 < R_N; k0 += 4) {
        v2f a = *(const v2f*)(cRow + k0 + kk);
        float b0 = rbf[(size_t)(k0 + kk    ) * D_N + dcol];
        float b1 = rbf[(size_t)(k0 + kk + 1) * D_N + dcol];
        v2f bv; bv.x = b0; bv.y = b1;
        v2f b2 = bv * bv;
        acc1 = wmma_f32_k4(a, bv, acc1);
        acc2 = wmma_f32_k4(a, b2, acc2);
    }
    #pragma unroll
    for (int j = 0; j < 8; ++j) {
        int row  = bm * 16 + j + hi * 8;
        float zv = z[(size_t)row * D_N + dcol];
        float grad = zv * zv * Csum[row] - 2.f * zv * acc1[j] + acc2[j];
        newAttn[(size_t)row * D_N + dcol] =
            fmaxf(attn[(size_t)row * D_N + dcol] - LAM_A * grad, 0.f);
    }
}

// ---------------------------------------------------------------------------
extern "C" void kernel_launch(void* const* d_in, const int* in_sizes, int n_in,
                              void* d_out, int out_size, void* d_ws, size_t ws_size,
                              hipStream_t stream)
{
    (void)in_sizes; (void)n_in; (void)out_size; (void)ws_size;
    const float* z      = (const float*)d_in[0];   // (B,D)
    const float* onehot = (const float*)d_in[1];   // (B,O)
    const float* attn   = (const float*)d_in[2];   // (B,D)
    const float* assoc  = (const float*)d_in[3];   // (B,R,O)
    const float* rbf    = (const float*)d_in[4];   // (R,D)

    float* out      = (float*)d_out;
    float* xout     = out;                               // (B,O)
    float* newAttn  = out + B_N * O_N;                   // (B,D)
    float* newAssoc = out + B_N * O_N + B_N * D_N;       // (B,R,O)

    float* ws    = (float*)d_ws;
    float* sArr  = ws;                    // B*R
    float* scArr = sArr  + B_N * R_N;     // B*R
    float* cArr  = scArr + B_N * R_N;     // B*R
    float* CsumV = cArr  + B_N * R_N;     // B
    float* Ab    = CsumV + B_N;           // B

    alcove_rowsum<<<B_N / 8, 256, 0, stream>>>(attn, z, Ab);
    alcove_rbf<<<(B_N / 16) * (R_N / 16) / 8, 256, 0, stream>>>(
        attn, z, rbf, Ab, sArr, scArr);
    alcove_assoc<<<B_N, 256, 0, stream>>>(
        assoc, onehot, sArr, scArr, xout, newAssoc, cArr, CsumV);
    alcove_attn<<<(B_N / 16) * (D_N / 16) / 8, 256, 0, stream>>>(
        cArr, rbf, z, attn, CsumV, newAttn);
}